// LovaszLoss_40458591928841
// MI455X (gfx1250) — compile-verified
//
#include <hip/hip_runtime.h>
#include <math.h>

// Lovasz hinge on flat binary labels.
// Algebraic reduction (see analysis): with e = 1 - logit*sign, sign = 2*label-1,
//   loss = relu(max(e)) * sum(labels) - sum(relu(e) * labels)
// -> pure streaming reduction, HBM-bound (128 MB @ 23.3 TB/s ~ 5.5 us).
// Block sums are reduced with V_WMMA_F32_16X16X4_F32 (B = ones => row sums).

typedef float v2f __attribute__((ext_vector_type(2)));
typedef float v8f __attribute__((ext_vector_type(8)));

#define BLOCK 256
#define NBLK_MAX 2048

// Sum 256 floats staged in sh[0..255] using WMMA f32 16x16x4 with B = ones.
// Must be executed by a full wave (threads 0..31 of the block), EXEC all 1s.
// D[m,n] = sum_k A[m,k]; chaining 4 chunks accumulates all 256 values into the
// 16 row sums. Each lane sums its 8 D VGPRs (rows 0-7 in lanes 0-15, rows 8-15
// in lanes 16-31); shfl_xor(16) merges the halves. Returns total in all lanes.
__device__ __forceinline__ float wmma_block_sum(const float* sh, int lane) {
  v2f ones; ones[0] = 1.0f; ones[1] = 1.0f;
  v8f acc = {};
#pragma unroll
  for (int c = 0; c < 4; ++c) {
    v2f a;
    a[0] = sh[c * 64 + lane];
    a[1] = sh[c * 64 + 32 + lane];
    acc = __builtin_amdgcn_wmma_f32_16x16x4_f32(
        /*neg_a=*/false, a, /*neg_b=*/false, ones,
        /*c_mod=*/(short)0, acc, /*reuse_a=*/false, /*reuse_b=*/false);
  }
  float s = acc[0] + acc[1] + acc[2] + acc[3] + acc[4] + acc[5] + acc[6] + acc[7];
  s += __shfl_xor(s, 16, 32);
  return s;
}

// Max of 256 floats staged in sh[0..255], executed by wave 0 (lanes 0..31).
__device__ __forceinline__ float wave_max_256(const float* sh, int lane) {
  float m = -INFINITY;
#pragma unroll
  for (int c = 0; c < 8; ++c) m = fmaxf(m, sh[c * 32 + lane]);
#pragma unroll
  for (int off = 16; off > 0; off >>= 1) m = fmaxf(m, __shfl_xor(m, off, 32));
  return m;
}

__global__ void lovasz_partial_kernel(const float* __restrict__ logits,
                                      const float* __restrict__ labels,
                                      float* __restrict__ pdot,
                                      float* __restrict__ plab,
                                      float* __restrict__ pmax,
                                      int P) {
  __shared__ float sh[3 * BLOCK];
  const int tid = blockIdx.x * blockDim.x + threadIdx.x;
  const int stride = gridDim.x * blockDim.x;

  float dsum = 0.0f;       // sum relu(e) * label
  float lsum = 0.0f;       // sum label
  float emax = -INFINITY;  // max e

  const int P4 = P >> 2;
  const float4* __restrict__ x4 = (const float4*)logits;
  const float4* __restrict__ l4 = (const float4*)labels;

  for (int i = tid; i < P4; i += stride) {
    float4 x = x4[i];
    float4 l = l4[i];
#define PROC(xx, ll)                          \
    {                                         \
      float s_ = 2.0f * (ll) - 1.0f;          \
      float e_ = fmaf(-(xx), s_, 1.0f);       \
      emax = fmaxf(emax, e_);                 \
      float r_ = fmaxf(e_, 0.0f);             \
      dsum = fmaf(r_, (ll), dsum);            \
      lsum += (ll);                           \
    }
    PROC(x.x, l.x) PROC(x.y, l.y) PROC(x.z, l.z) PROC(x.w, l.w)
  }
  // scalar tail (P not a multiple of 4)
  for (int i = (P4 << 2) + tid; i < P; i += stride) {
    float x = logits[i];
    float l = labels[i];
    PROC(x, l)
  }
#undef PROC

  sh[threadIdx.x] = dsum;
  sh[BLOCK + threadIdx.x] = lsum;
  sh[2 * BLOCK + threadIdx.x] = emax;
  __syncthreads();

  if (threadIdx.x < 32) {
    const int lane = threadIdx.x;
    float td = wmma_block_sum(sh, lane);
    float tl = wmma_block_sum(sh + BLOCK, lane);
    float tm = wave_max_256(sh + 2 * BLOCK, lane);
    if (lane == 0) {
      pdot[blockIdx.x] = td;
      plab[blockIdx.x] = tl;
      pmax[blockIdx.x] = tm;
    }
  }
}

__global__ void lovasz_final_kernel(const float* __restrict__ pdot,
                                    const float* __restrict__ plab,
                                    const float* __restrict__ pmax,
                                    int nblk,
                                    float* __restrict__ out) {
  __shared__ float sh[3 * BLOCK];
  float d = 0.0f, l = 0.0f, m = -INFINITY;
  for (int i = threadIdx.x; i < nblk; i += BLOCK) {
    d += pdot[i];
    l += plab[i];
    m = fmaxf(m, pmax[i]);
  }
  sh[threadIdx.x] = d;
  sh[BLOCK + threadIdx.x] = l;
  sh[2 * BLOCK + threadIdx.x] = m;
  __syncthreads();

  if (threadIdx.x < 32) {
    const int lane = threadIdx.x;
    float td = wmma_block_sum(sh, lane);
    float tl = wmma_block_sum(sh + BLOCK, lane);
    float tm = wave_max_256(sh + 2 * BLOCK, lane);
    if (lane == 0) {
      out[0] = fmaxf(tm, 0.0f) * tl - td;  // relu(max_e)*gts - dot
    }
  }
}

extern "C" void kernel_launch(void* const* d_in, const int* in_sizes, int n_in,
                              void* d_out, int out_size, void* d_ws, size_t ws_size,
                              hipStream_t stream) {
  const float* logits = (const float*)d_in[0];
  const float* labels = (const float*)d_in[1];
  const int P = in_sizes[0];
  float* out = (float*)d_out;

  float* pdot = (float*)d_ws;
  float* plab = pdot + NBLK_MAX;
  float* pmax = plab + NBLK_MAX;

  int nblk = NBLK_MAX;
  int need = ((P >> 2) + BLOCK - 1) / BLOCK;
  if (need < 1) need = 1;
  if (nblk > need) nblk = need;

  lovasz_partial_kernel<<<nblk, BLOCK, 0, stream>>>(logits, labels, pdot, plab, pmax, P);
  lovasz_final_kernel<<<1, BLOCK, 0, stream>>>(pdot, plab, pmax, nblk, out);
}